// PointerGenerator_20976620274326
// MI455X (gfx1250) — compile-verified
//
#include <hip/hip_runtime.h>
#include <math.h>

// Problem constants (from the reference)
#define B_   32
#define S_   400
#define H_   256
#define E_   300
#define V_   50000
#define T_   20
#define OOV_ 50
#define TWOH_ 512
#define FOURH_ 1024
#define VEXT_ (V_ + OOV_)

typedef __attribute__((ext_vector_type(2))) float v2f;
typedef __attribute__((ext_vector_type(8))) float v8f;

// ---------------------------------------------------------------------------
// WMMA f32 16x16x4 micro-tile accumulate over a K-segment.
//  A: per-lane row pointer (row = mtile*16 + (lane&15)); lanes 16-31 carry K+2/K+3.
//  B: row-major [K, ldb], col = ntile*16 + (lane&15).
//  Emits v_wmma_f32_16x16x4_f32 (fp32 matrix pipe, preserves reference precision).
// ---------------------------------------------------------------------------
__device__ __forceinline__ v8f wmma_seg_f32(v8f acc,
                                            const float* __restrict__ Arow,
                                            const float* __restrict__ B, int ldb,
                                            int K, int col, int kHalf) {
  for (int k = 0; k < K; k += 4) {
    v2f a, b;
    a[0] = Arow[k + kHalf];
    a[1] = Arow[k + kHalf + 1];
    b[0] = B[(size_t)(k + kHalf) * ldb + col];
    b[1] = B[(size_t)(k + kHalf + 1) * ldb + col];
    if ((k & 60) == 0)  // prefetch next weight block (global_prefetch_b8)
      __builtin_prefetch(&B[(size_t)(k + 64) * ldb + col], 0, 0);
    acc = __builtin_amdgcn_wmma_f32_16x16x4_f32(false, a, false, b,
                                                (short)0, acc, false, false);
  }
  return acc;
}

__device__ __forceinline__ float sigm(float x) { return 1.0f / (1.0f + expf(-x)); }

// ---------------------------------------------------------------------------
// Init per-launch state: h = h0, c = c0, cov = 0
// ---------------------------------------------------------------------------
__global__ void init_state(const float* __restrict__ h0, const float* __restrict__ c0,
                           float* __restrict__ h, float* __restrict__ c,
                           float* __restrict__ cov) {
  int i = blockIdx.x * blockDim.x + threadIdx.x;
  if (i < B_ * H_) { h[i] = h0[i]; c[i] = c0[i]; }
  if (i < B_ * S_) cov[i] = 0.0f;
}

// ---------------------------------------------------------------------------
// enc_feat = enc_hidden(12800x512) @ W_enc(512x256)   — WMMA
// grid(4, 800) block 128 (4 waves, each one 16x16 tile)
// ---------------------------------------------------------------------------
__global__ void enc_feat_gemm(const float* __restrict__ enc_hidden,
                              const float* __restrict__ W_enc,
                              float* __restrict__ enc_feat) {
  int wid = threadIdx.x >> 5, lane = threadIdx.x & 31;
  int ntile = blockIdx.x * 4 + wid;   // 16 N tiles
  int mtile = blockIdx.y;             // 800 M tiles
  int row = mtile * 16 + (lane & 15);
  int col = ntile * 16 + (lane & 15);
  int kHalf = (lane >> 4) * 2;
  v8f acc = {};
  acc = wmma_seg_f32(acc, enc_hidden + (size_t)row * TWOH_, W_enc, H_, TWOH_, col, kHalf);
  int rbase = mtile * 16 + (lane >> 4) * 8;
  for (int j = 0; j < 8; ++j)
    enc_feat[(size_t)(rbase + j) * H_ + col] = acc[j];
}

// ---------------------------------------------------------------------------
// gates = [emb | h] @ [W_ih ; W_hh] + b_lstm   (M=32, N=1024, K=300+256) — WMMA
// emb gathered inline from embedding[dec_input[:,t]].
// grid(16, 2) block 128
// ---------------------------------------------------------------------------
__global__ void lstm_gates(const float* __restrict__ embedding,
                           const int* __restrict__ dec_input,
                           const float* __restrict__ h,
                           const float* __restrict__ W_ih,
                           const float* __restrict__ W_hh,
                           const float* __restrict__ b_lstm,
                           float* __restrict__ gates, int t) {
  int wid = threadIdx.x >> 5, lane = threadIdx.x & 31;
  int ntile = blockIdx.x * 4 + wid;   // 64 N tiles
  int mtile = blockIdx.y;             // 2 M tiles
  int row = mtile * 16 + (lane & 15);
  int col = ntile * 16 + (lane & 15);
  int kHalf = (lane >> 4) * 2;
  int tok = dec_input[row * T_ + t];
  v8f acc = {};
  acc = wmma_seg_f32(acc, embedding + (size_t)tok * E_, W_ih, FOURH_, E_, col, kHalf);
  acc = wmma_seg_f32(acc, h + (size_t)row * H_, W_hh, FOURH_, H_, col, kHalf);
  int rbase = mtile * 16 + (lane >> 4) * 8;
  float bv = b_lstm[col];
  for (int j = 0; j < 8; ++j)
    gates[(size_t)(rbase + j) * FOURH_ + col] = acc[j] + bv;
}

// ---------------------------------------------------------------------------
// LSTM pointwise: c,h update in place. 8192 threads.
// ---------------------------------------------------------------------------
__global__ void lstm_update(const float* __restrict__ gates,
                            float* __restrict__ h, float* __restrict__ c) {
  int idx = blockIdx.x * blockDim.x + threadIdx.x;
  if (idx >= B_ * H_) return;
  int b = idx >> 8, j = idx & (H_ - 1);
  const float* g = gates + (size_t)b * FOURH_;
  float gi = g[j], gf = g[H_ + j], gg = g[2 * H_ + j], go = g[3 * H_ + j];
  float cn = sigm(gf) * c[idx] + sigm(gi) * tanhf(gg);
  float hn = sigm(go) * tanhf(cn);
  c[idx] = cn;
  h[idx] = hn;
}

// ---------------------------------------------------------------------------
// Generic M=32 WMMA GEMM: D = A(32xK) @ B(KxN) [+ bias]
// grid(ceil(ntiles/4), 2) block 128. Used for h@W_dec and out1@W_out2.
// ---------------------------------------------------------------------------
__global__ void gemm_a32(const float* __restrict__ A, int lda,
                         const float* __restrict__ Bm, int ldb,
                         const float* __restrict__ bias,
                         float* __restrict__ D, int ldd, int K, int ntiles) {
  int wid = threadIdx.x >> 5, lane = threadIdx.x & 31;
  int ntile = blockIdx.x * 4 + wid;
  if (ntile >= ntiles) return;  // wave-uniform guard (EXEC stays all-ones)
  int mtile = blockIdx.y;
  int row = mtile * 16 + (lane & 15);
  int col = ntile * 16 + (lane & 15);
  int kHalf = (lane >> 4) * 2;
  v8f acc = {};
  acc = wmma_seg_f32(acc, A + (size_t)row * lda, Bm, ldb, K, col, kHalf);
  int rbase = mtile * 16 + (lane >> 4) * 8;
  float bv = bias ? bias[col] : 0.0f;
  for (int j = 0; j < 8; ++j)
    D[(size_t)(rbase + j) * ldd + col] = acc[j] + bv;
}

// ---------------------------------------------------------------------------
// out1 = [h | context] @ W_out1 + b_out1   (M=32, N=256, K=256+512) — WMMA
// grid(4, 2) block 128
// ---------------------------------------------------------------------------
__global__ void out1_gemm(const float* __restrict__ h,
                          const float* __restrict__ context,
                          const float* __restrict__ W_out1,
                          const float* __restrict__ b_out1,
                          float* __restrict__ out1) {
  int wid = threadIdx.x >> 5, lane = threadIdx.x & 31;
  int ntile = blockIdx.x * 4 + wid;   // 16 N tiles
  int mtile = blockIdx.y;             // 2 M tiles
  int row = mtile * 16 + (lane & 15);
  int col = ntile * 16 + (lane & 15);
  int kHalf = (lane >> 4) * 2;
  v8f acc = {};
  acc = wmma_seg_f32(acc, h + (size_t)row * H_, W_out1, H_, H_, col, kHalf);
  acc = wmma_seg_f32(acc, context + (size_t)row * TWOH_,
                     W_out1 + (size_t)H_ * H_, H_, TWOH_, col, kHalf);
  int rbase = mtile * 16 + (lane >> 4) * 8;
  float bv = b_out1[col];
  for (int j = 0; j < 8; ++j)
    out1[(size_t)(rbase + j) * H_ + col] = acc[j] + bv;
}

// ---------------------------------------------------------------------------
// e[b,s] = v_attn . tanh(enc_feat[b,s,:] + dec_feat[b,:] + cov[b,s]*w_cov)
// one wave per (b,s) row; wave32 butterfly reduce. grid 1600, block 256.
// ---------------------------------------------------------------------------
__global__ void attn_scores(const float* __restrict__ enc_feat,
                            const float* __restrict__ dec_feat,
                            const float* __restrict__ cov,
                            const float* __restrict__ w_cov,
                            const float* __restrict__ v_attn,
                            const unsigned char* __restrict__ mask,
                            float* __restrict__ e) {
  int wid = threadIdx.x >> 5, lane = threadIdx.x & 31;
  int rowIdx = blockIdx.x * 8 + wid;          // B_*S_ rows
  if (rowIdx >= B_ * S_) return;
  int b = rowIdx / S_;
  float cv = cov[rowIdx];
  const float* ef = enc_feat + (size_t)rowIdx * H_;
  const float* df = dec_feat + (size_t)b * H_;
  float acc = 0.0f;
  for (int j = 0; j < 8; ++j) {
    int d = lane + j * 32;                    // coalesced across the wave
    float xv = ef[d] + df[d] + cv * w_cov[d];
    acc += tanhf(xv) * v_attn[d];
  }
  for (int off = 16; off; off >>= 1) acc += __shfl_xor(acc, off, 32);
  if (lane == 0) e[rowIdx] = mask[rowIdx] ? acc : -1e9f;
}

// ---------------------------------------------------------------------------
// Per-b: softmax over S, context = attn @ enc_hidden, cov += attn,
// write attn/cov time-slices. grid 32, block 256.
// ---------------------------------------------------------------------------
__global__ void softmax_ctx(const float* __restrict__ e,
                            const float* __restrict__ enc_hidden,
                            float* __restrict__ attn,
                            float* __restrict__ cov,
                            float* __restrict__ context,
                            float* __restrict__ attns_out,
                            float* __restrict__ covs_out, int t) {
  __shared__ float red[256];
  __shared__ float a_sh[S_];
  int b = blockIdx.x, tid = threadIdx.x;
  const float* eb = e + (size_t)b * S_;

  float m = -3.0e38f;
  for (int s = tid; s < S_; s += 256) m = fmaxf(m, eb[s]);
  red[tid] = m; __syncthreads();
  for (int off = 128; off; off >>= 1) {
    if (tid < off) red[tid] = fmaxf(red[tid], red[tid + off]);
    __syncthreads();
  }
  m = red[0]; __syncthreads();

  float sum = 0.0f;
  for (int s = tid; s < S_; s += 256) { float ex = expf(eb[s] - m); a_sh[s] = ex; sum += ex; }
  red[tid] = sum; __syncthreads();
  for (int off = 128; off; off >>= 1) {
    if (tid < off) red[tid] += red[tid + off];
    __syncthreads();
  }
  float inv = 1.0f / red[0];

  for (int s = tid; s < S_; s += 256) {
    float a = a_sh[s] * inv;
    a_sh[s] = a;
    attn[(size_t)b * S_ + s] = a;
    float cn = cov[(size_t)b * S_ + s] + a;
    cov[(size_t)b * S_ + s] = cn;
    attns_out[(size_t)(b * S_ + s) * T_ + t] = a;
    covs_out[(size_t)(b * S_ + s) * T_ + t] = cn;
  }
  __syncthreads();

  // context[b,d] = sum_s attn[s] * enc_hidden[b,s,d]   (d coalesced)
  for (int d = tid; d < TWOH_; d += 256) {
    const float* eh = enc_hidden + (size_t)b * S_ * TWOH_ + d;
    float acc = 0.0f;
    for (int s = 0; s < S_; ++s) acc += a_sh[s] * eh[(size_t)s * TWOH_];
    context[(size_t)b * TWOH_ + d] = acc;
  }
}

// ---------------------------------------------------------------------------
// Per-b: p_gen, vocab softmax scaled by p_gen, OOV zeros, scatter-add copy
// probabilities into finals[b,:,t]. grid 32, block 256.
// ---------------------------------------------------------------------------
__global__ void final_dist(const float* __restrict__ logits,
                           const float* __restrict__ h,
                           const float* __restrict__ context,
                           const float* __restrict__ attn,
                           const float* __restrict__ embedding,
                           const int* __restrict__ dec_input,
                           const int* __restrict__ enc_input_ext,
                           const float* __restrict__ w_h,
                           const float* __restrict__ w_s,
                           const float* __restrict__ w_x,
                           const float* __restrict__ b_x,
                           float* __restrict__ finals, int t) {
  __shared__ float red[256];
  __shared__ float sh_pgen;
  int b = blockIdx.x, tid = threadIdx.x;

  // p_gen = sigmoid(ctx.w_h + h.w_s + emb.w_x + b_x)
  float acc = 0.0f;
  for (int d = tid; d < TWOH_; d += 256) acc += context[(size_t)b * TWOH_ + d] * w_h[d];
  for (int d = tid; d < H_; d += 256)    acc += h[(size_t)b * H_ + d] * w_s[d];
  int tok = dec_input[b * T_ + t];
  for (int d = tid; d < E_; d += 256)    acc += embedding[(size_t)tok * E_ + d] * w_x[d];
  red[tid] = acc; __syncthreads();
  for (int off = 128; off; off >>= 1) {
    if (tid < off) red[tid] += red[tid + off];
    __syncthreads();
  }
  if (tid == 0) sh_pgen = sigm(red[0] + b_x[0]);
  __syncthreads();
  float pg = sh_pgen;

  const float* lg = logits + (size_t)b * V_;
  float m = -3.0e38f;
  for (int v = tid; v < V_; v += 256) m = fmaxf(m, lg[v]);
  red[tid] = m; __syncthreads();
  for (int off = 128; off; off >>= 1) {
    if (tid < off) red[tid] = fmaxf(red[tid], red[tid + off]);
    __syncthreads();
  }
  m = red[0]; __syncthreads();

  float sum = 0.0f;
  for (int v = tid; v < V_; v += 256) sum += expf(lg[v] - m);
  red[tid] = sum; __syncthreads();
  for (int off = 128; off; off >>= 1) {
    if (tid < off) red[tid] += red[tid + off];
    __syncthreads();
  }
  float scale = pg / red[0];

  float* fo = finals + (size_t)b * VEXT_ * T_ + t;   // finals[b, v, t]
  for (int v = tid; v < V_; v += 256)       fo[(size_t)v * T_] = expf(lg[v] - m) * scale;
  for (int v = V_ + tid; v < VEXT_; v += 256) fo[(size_t)v * T_] = 0.0f;
  __syncthreads();

  float wpg = 1.0f - pg;
  for (int s = tid; s < S_; s += 256) {
    int ext = enc_input_ext[b * S_ + s];
    atomicAdd(&fo[(size_t)ext * T_], wpg * attn[(size_t)b * S_ + s]);
  }
}

// ---------------------------------------------------------------------------
// Host launch
// ---------------------------------------------------------------------------
extern "C" void kernel_launch(void* const* d_in, const int* in_sizes, int n_in,
                              void* d_out, int out_size, void* d_ws, size_t ws_size,
                              hipStream_t stream) {
  const float* enc_hidden   = (const float*)d_in[0];
  const float* h0           = (const float*)d_in[1];
  const float* c0           = (const float*)d_in[2];
  const float* embedding    = (const float*)d_in[3];
  const float* W_enc        = (const float*)d_in[4];
  const float* W_dec        = (const float*)d_in[5];
  const float* w_cov        = (const float*)d_in[6];
  const float* v_attn       = (const float*)d_in[7];
  const float* W_ih         = (const float*)d_in[8];
  const float* W_hh         = (const float*)d_in[9];
  const float* b_lstm       = (const float*)d_in[10];
  const float* W_out1       = (const float*)d_in[11];
  const float* b_out1       = (const float*)d_in[12];
  const float* W_out2       = (const float*)d_in[13];
  const float* b_out2       = (const float*)d_in[14];
  const float* w_h          = (const float*)d_in[15];
  const float* w_s          = (const float*)d_in[16];
  const float* w_x          = (const float*)d_in[17];
  const float* b_x          = (const float*)d_in[18];
  const int*   dec_input    = (const int*)d_in[19];
  const int*   enc_input_ext= (const int*)d_in[20];
  const unsigned char* mask = (const unsigned char*)d_in[21];  // bool array

  float* finals    = (float*)d_out;                                   // [B,VEXT,T]
  float* attns_out = finals + (size_t)B_ * VEXT_ * T_;                // [B,S,T]
  float* covs_out  = attns_out + (size_t)B_ * S_ * T_;                // [B,S,T]

  // Workspace carve (~20 MB of fp32)
  float* ws       = (float*)d_ws;
  float* enc_feat = ws;                         // B*S*H   = 3,276,800
  float* h        = enc_feat + (size_t)B_ * S_ * H_;   // 8192
  float* c        = h + B_ * H_;                       // 8192
  float* cov      = c + B_ * H_;                       // 12800
  float* gates    = cov + B_ * S_;                     // 32768
  float* dec_feat = gates + B_ * FOURH_;               // 8192
  float* e        = dec_feat + B_ * H_;                // 12800
  float* attn     = e + B_ * S_;                       // 12800
  float* context  = attn + B_ * S_;                    // 16384
  float* out1     = context + B_ * TWOH_;              // 8192
  float* logits   = out1 + B_ * H_;                    // 1,600,000

  init_state<<<(B_ * S_ + 255) / 256, 256, 0, stream>>>(h0, c0, h, c, cov);
  enc_feat_gemm<<<dim3(4, (B_ * S_) / 16), 128, 0, stream>>>(enc_hidden, W_enc, enc_feat);

  const int vocab_ntiles = V_ / 16;  // 3125
  for (int t = 0; t < T_; ++t) {
    lstm_gates<<<dim3(16, 2), 128, 0, stream>>>(embedding, dec_input, h,
                                                W_ih, W_hh, b_lstm, gates, t);
    lstm_update<<<(B_ * H_ + 255) / 256, 256, 0, stream>>>(gates, h, c);
    gemm_a32<<<dim3(4, 2), 128, 0, stream>>>(h, H_, W_dec, H_, nullptr,
                                             dec_feat, H_, H_, H_ / 16);
    attn_scores<<<(B_ * S_) / 8, 256, 0, stream>>>(enc_feat, dec_feat, cov,
                                                   w_cov, v_attn, mask, e);
    softmax_ctx<<<B_, 256, 0, stream>>>(e, enc_hidden, attn, cov, context,
                                        attns_out, covs_out, t);
    out1_gemm<<<dim3(4, 2), 128, 0, stream>>>(h, context, W_out1, b_out1, out1);
    gemm_a32<<<dim3((vocab_ntiles + 3) / 4, 2), 128, 0, stream>>>(
        out1, H_, W_out2, V_, b_out2, logits, V_, H_, vocab_ntiles);
    final_dist<<<B_, 256, 0, stream>>>(logits, h, context, attn, embedding,
                                       dec_input, enc_input_ext,
                                       w_h, w_s, w_x, b_x, finals, t);
  }
}